// Contrastiveloss_9500467658849
// MI455X (gfx1250) — compile-verified
//
#include <hip/hip_runtime.h>
#include <math.h>

typedef float v2f __attribute__((ext_vector_type(2)));
typedef float v8f __attribute__((ext_vector_type(8)));

#define N_ROWS 8192
#define DIM    128
#define KSTEPS (DIM / 4)                  // 32 WMMA f32 16x16x4 steps per 16x16 tile
#define JSPLIT 4                          // j-range slices per i-tile (occupancy)
#define JSLICE (N_ROWS / JSPLIT)          // 2048 rows per slice
#define INV_TAU_LOG2E 14.426950408889634f // log2(e)/tau, tau=0.1

// ---------------- Kernel 1: rnorm[i] = 1/||x_i|| ----------------
__global__ void cl_rnorm_kernel(const float* __restrict__ x,
                                float* __restrict__ rnorm) {
    int i = blockIdx.x * blockDim.x + threadIdx.x;
    if (i >= N_ROWS) return;
    const float4* xr = (const float4*)(x + (size_t)i * DIM);
    float s = 0.f;
#pragma unroll
    for (int k = 0; k < DIM / 4; ++k) {
        float4 v = xr[k];
        s += v.x * v.x + v.y * v.y + v.z * v.z + v.w * v.w;
    }
    rnorm[i] = rsqrtf(s);
}

// ---- Kernel 2: fused S-tile = x̂ x̂ᵀ (fp32 WMMA) + exp + masked partial row sums ----
// 2048 waves: wave -> (i-tile = gw>>2, j-slice = gw&3). Each wave:
//   * preloads its 16-row A fragment (normalization folded in),
//   * per 16-col j-tile: burst-loads the whole B fragment into registers,
//     then runs the 32-step WMMA chain on two interleaved accumulators,
//   * applies exp2 + label mask, accumulates per-row top/down partials,
//   * writes partials (per slice) for the final combine kernel.
__global__ void __launch_bounds__(256, 1)
cl_simloss_kernel(const float* __restrict__ x,
                  const int*   __restrict__ y32,   // low dwords of int64 labels (stride 2)
                  const float* __restrict__ rnorm,
                  float* __restrict__ topp,        // [JSPLIT][N_ROWS]
                  float* __restrict__ downp) {     // [JSPLIT][N_ROWS]
    const int lane  = threadIdx.x & 31;
    const int wave  = threadIdx.x >> 5;
    const int half  = lane >> 4;   // K-pair select: 0 -> {4s,4s+1}, 1 -> {4s+2,4s+3}
    const int lm    = lane & 15;
    const int gw    = blockIdx.x * 8 + wave;   // 0..2047
    const int tile  = gw >> 2;                 // 0..511
    const int slice = gw & (JSPLIT - 1);
    const int i0    = tile * 16;
    const int jbeg  = slice * JSLICE;
    const int jend  = jbeg + JSLICE;

    // Preload A fragment for all 32 K-steps, normalization folded in.
    const int    rA  = i0 + lm;
    const float  rnA = rnorm[rA];
    const float* xA  = x + (size_t)rA * DIM + half * 2;
    v2f a[KSTEPS];
#pragma unroll
    for (int s = 0; s < KSTEPS; ++s) {
        v2f t = *(const v2f*)(xA + s * 4);
        t.x *= rnA; t.y *= rnA;
        a[s] = t;
    }

    // Labels of my 8 output rows (C VGPR r -> global row i0 + r + 8*half).
    int yi[8];
#pragma unroll
    for (int r = 0; r < 8; ++r) yi[r] = y32[2 * (i0 + r + 8 * half)];

    float top[8], down[8];
#pragma unroll
    for (int r = 0; r < 8; ++r) { top[r] = 0.f; down[r] = 0.f; }

    for (int j0 = jbeg; j0 < jend; j0 += 16) {
        const int    rB = j0 + lm;             // same gather pattern as A (S symmetric)
        const float* xB = x + (size_t)rB * DIM + half * 2;

        // Burst-load the full B fragment first (32 independent b64 loads).
        v2f braw[KSTEPS];
#pragma unroll
        for (int s = 0; s < KSTEPS; ++s) braw[s] = *(const v2f*)(xB + s * 4);
        const float rnB = rnorm[rB];
        const int   yj  = y32[2 * rB];

        // Two interleaved accumulators break the WMMA->WMMA dependency chain.
        v8f c0 = {}, c1 = {};
#pragma unroll
        for (int s = 0; s < KSTEPS; s += 2) {
            v2f b0; b0.x = braw[s].x * rnB;     b0.y = braw[s].y * rnB;
            c0 = __builtin_amdgcn_wmma_f32_16x16x4_f32(
                     false, a[s], false, b0, (short)0, c0, false, false);
            v2f b1; b1.x = braw[s + 1].x * rnB; b1.y = braw[s + 1].y * rnB;
            c1 = __builtin_amdgcn_wmma_f32_16x16x4_f32(
                     false, a[s + 1], false, b1, (short)0, c1, false, false);
        }

#pragma unroll
        for (int r = 0; r < 8; ++r) {
            float e = exp2f((c0[r] + c1[r]) * INV_TAU_LOG2E);
            down[r] += e;
            top[r]  += (yi[r] == yj) ? e : 0.f;
        }
    }

    // Reduce across the 16 lanes of each half (xor masks 1..8 stay in-half).
#pragma unroll
    for (int r = 0; r < 8; ++r) {
        float t = top[r], d = down[r];
#pragma unroll
        for (int m = 8; m >= 1; m >>= 1) {
            t += __shfl_xor(t, m, 32);
            d += __shfl_xor(d, m, 32);
        }
        top[r] = t; down[r] = d;
    }

    if (lm == 0) {
#pragma unroll
        for (int r = 0; r < 8; ++r) {
            int row = i0 + r + 8 * half;
            topp [slice * N_ROWS + row] = top[r];
            downp[slice * N_ROWS + row] = down[r];
        }
    }
}

// ---- Kernel 3: combine slice partials, take logs, deterministic mean ----
__global__ void cl_reduce_kernel(const float* __restrict__ topp,
                                 const float* __restrict__ downp,
                                 float* __restrict__ out) {
    __shared__ float red[256];
    float s = 0.f;
    for (int i = threadIdx.x; i < N_ROWS; i += 256) {
        float t = 0.f, d = 0.f;
#pragma unroll
        for (int p = 0; p < JSPLIT; ++p) {
            t += topp [p * N_ROWS + i];
            d += downp[p * N_ROWS + i];
        }
        s += logf(d) - logf(t);   // -log(top/down)
    }
    red[threadIdx.x] = s;
    __syncthreads();
    for (int off = 128; off > 0; off >>= 1) {
        if ((int)threadIdx.x < off) red[threadIdx.x] += red[threadIdx.x + off];
        __syncthreads();
    }
    if (threadIdx.x == 0) out[0] = red[0] * (1.0f / N_ROWS);
}

extern "C" void kernel_launch(void* const* d_in, const int* in_sizes, int n_in,
                              void* d_out, int out_size, void* d_ws, size_t ws_size,
                              hipStream_t stream) {
    const float* x   = (const float*)d_in[0];   // [8192,128] f32
    const int*   y32 = (const int*)d_in[1];     // int64 labels; low dwords (LE)
    float* out = (float*)d_out;                 // scalar loss

    // Workspace: rnorm [N] | topp [JSPLIT*N] | downp [JSPLIT*N]  -> 288 KB
    float* rnorm = (float*)d_ws;
    float* topp  = rnorm + N_ROWS;
    float* downp = topp + JSPLIT * N_ROWS;

    cl_rnorm_kernel<<<N_ROWS / 256, 256, 0, stream>>>(x, rnorm);
    cl_simloss_kernel<<<(512 * JSPLIT) / 8, 256, 0, stream>>>(x, y32, rnorm, topp, downp);
    cl_reduce_kernel<<<1, 256, 0, stream>>>(topp, downp, out);
}